// SignedLayer_67585605369883
// MI455X (gfx1250) — compile-verified
//
#include <hip/hip_runtime.h>
#include <hip/hip_bf16.h>

typedef float v2f __attribute__((ext_vector_type(2)));
typedef float v8f __attribute__((ext_vector_type(8)));

#define DIM 64

// ------------------------------------------------------------- zero ints
__global__ void sl_zeroi_kernel(int* p, int n) {
    int i = blockIdx.x * blockDim.x + threadIdx.x;
    if (i < n) p[i] = 0;
}

// ------------------------------------------------- tiny prep: derived matrices
// P = W2 @ W1 ; Q = W1 - W2 @ W2 ; W2m = -W2
// bias_r[n] = b1[n]-b2[n]
// bias_i[n] = b1[n]+b2[n] + sum_k (b1[k]-b2[k]) * W2[n][k]
__global__ void sl_prep_kernel(const float* __restrict__ W1,
                               const float* __restrict__ W2,
                               const float* __restrict__ b1,
                               const float* __restrict__ b2,
                               float* __restrict__ W2m,
                               float* __restrict__ P,
                               float* __restrict__ Q,
                               float* __restrict__ bias_r,
                               float* __restrict__ bias_i) {
    int i = blockIdx.x;    // row 0..63
    int j = threadIdx.x;   // col 0..63
    float sp = 0.f, sq = 0.f;
    for (int t = 0; t < DIM; ++t) {
        float w2 = W2[i * DIM + t];
        sp = fmaf(w2, W1[t * DIM + j], sp);
        sq = fmaf(w2, W2[t * DIM + j], sq);
    }
    P[i * DIM + j] = sp;
    Q[i * DIM + j] = W1[i * DIM + j] - sq;
    W2m[i * DIM + j] = -W2[i * DIM + j];
    if (i == 0) {
        float s = 0.f;
        for (int k = 0; k < DIM; ++k)
            s = fmaf(b1[k] - b2[k], W2[j * DIM + k], s);
        bias_r[j] = b1[j] - b2[j];
        bias_i[j] = b1[j] + b2[j] + s;
    }
}

// ----------------------------------------------------- CSR build: count by dst
__global__ void sl_count_kernel(const int* __restrict__ dst, int* __restrict__ cnt,
                                int n_edges) {
    int e = blockIdx.x * blockDim.x + threadIdx.x;
    if (e < n_edges) atomicAdd(&cnt[dst[e]], 1);
}

// ------------------------------------- CSR build: exclusive scan (single block)
__global__ void sl_scan_kernel(const int* __restrict__ cnt, int* __restrict__ offs,
                               int* __restrict__ cursor, int n) {
    __shared__ int sh[1024];
    __shared__ int s_carry;
    int tid = threadIdx.x;
    if (tid == 0) s_carry = 0;
    __syncthreads();
    for (int base = 0; base < n; base += 1024) {
        int i = base + tid;
        int v = (i < n) ? cnt[i] : 0;
        sh[tid] = v;
        __syncthreads();
        for (int off = 1; off < 1024; off <<= 1) {
            int t = (tid >= off) ? sh[tid - off] : 0;
            __syncthreads();
            sh[tid] += t;
            __syncthreads();
        }
        int incl = sh[tid];
        int carry = s_carry;
        if (i < n) {
            int excl = carry + incl - v;
            offs[i] = excl;
            cursor[i] = excl;
        }
        __syncthreads();
        if (tid == 1023) s_carry = carry + incl;
        __syncthreads();
    }
    if (tid == 0) offs[n] = s_carry;
}

// -------------------------------------------- CSR build: scatter edge indices
__global__ void sl_scatter_kernel(const int* __restrict__ dst, int* __restrict__ cursor,
                                  int* __restrict__ eidx, int n_edges) {
    int e = blockIdx.x * blockDim.x + threadIdx.x;
    if (e < n_edges) {
        int pos = atomicAdd(&cursor[dst[e]], 1);
        eidx[pos] = e;
    }
}

// -------------------------------------------------- gather-accumulate (dominant)
// One wave per node. Lanes cooperatively fetch 32 edge records (coalesced),
// then each edge's src/coefficients are shfl-broadcast and all 32 lanes
// FMA-accumulate the node's 64-dim complex row in registers. One store, no
// float atomics.
__global__ void sl_gather_kernel(const float* __restrict__ h_real,
                                 const float* __restrict__ h_imag,
                                 const float* __restrict__ dnode,
                                 const float* __restrict__ w_real,
                                 const float* __restrict__ w_imag,
                                 const int* __restrict__ src,
                                 const int* __restrict__ eidx,
                                 const int* __restrict__ offs,
                                 float* __restrict__ z_real,
                                 float* __restrict__ z_imag,
                                 int n_nodes) {
    int wave = threadIdx.x >> 5;
    int lane = threadIdx.x & 31;
    int n = blockIdx.x * (blockDim.x >> 5) + wave;
    if (n >= n_nodes) return;

    int start = offs[n];
    int end = offs[n + 1];
    float dn = dnode[n];

    float ar0 = 0.f, ar1 = 0.f, ai0 = 0.f, ai1 = 0.f;

    for (int base = start; base < end; base += 32) {
        int cnt = min(32, end - base);
        int j = base + lane;
        int s = 0;
        float er = 0.f, ei = 0.f;
        if (j < end) {
            int eid = eidx[j];
            s = src[eid];
            float c = dn * dnode[s];
            er = c * w_real[eid];
            ei = c * w_imag[eid];
        }
        for (int l = 0; l < cnt; ++l) {
            int   ss = __shfl(s, l);
            float cr = __shfl(er, l);
            float ci = __shfl(ei, l);
            v2f hr = *(const v2f*)(h_real + (size_t)ss * DIM + 2 * lane);
            v2f hi = *(const v2f*)(h_imag + (size_t)ss * DIM + 2 * lane);
            ar0 = fmaf(cr, hr.x, fmaf(-ci, hi.x, ar0));
            ar1 = fmaf(cr, hr.y, fmaf(-ci, hi.y, ar1));
            ai0 = fmaf(ci, hr.x, fmaf(cr, hi.x, ai0));
            ai1 = fmaf(ci, hr.y, fmaf(cr, hi.y, ai1));
        }
    }

    v2f zr = {ar0, ar1};
    v2f zi = {ai0, ai1};
    *(v2f*)(z_real + (size_t)n * DIM + 2 * lane) = zr;
    *(v2f*)(z_imag + (size_t)n * DIM + 2 * lane) = zi;
}

// --------------------------------------------------------------- WMMA GEMM pass
// zr = z_real@W1^T + z_imag@W2m^T + bias_r
// zi = z_real@P^T  + z_imag@Q^T   + bias_i
// One wave handles 16 nodes. V_WMMA_F32_16X16X4_F32, K looped 16x (K=64).
// nt loop is kept rolled (#pragma unroll 1): full unroll spills to scratch.
__global__ void sl_gemm_kernel(const float* __restrict__ z_real,
                               const float* __restrict__ z_imag,
                               const float* __restrict__ W1,
                               const float* __restrict__ W2m,
                               const float* __restrict__ P,
                               const float* __restrict__ Q,
                               const float* __restrict__ bias_r,
                               const float* __restrict__ bias_i,
                               float* __restrict__ out_zr,
                               float* __restrict__ out_zi,
                               int n_nodes) {
    __shared__ float lW[4 * DIM * DIM];  // 64 KB: W1 | W2m | P | Q

    for (int i = threadIdx.x; i < DIM * DIM; i += blockDim.x) {
        lW[0 * DIM * DIM + i] = W1[i];
        lW[1 * DIM * DIM + i] = W2m[i];
        lW[2 * DIM * DIM + i] = P[i];
        lW[3 * DIM * DIM + i] = Q[i];
    }
    __syncthreads();

    int wave = threadIdx.x >> 5;
    int lane = threadIdx.x & 31;
    int row0 = (blockIdx.x * (blockDim.x >> 5) + wave) * 16;
    if (row0 >= n_nodes) return;  // wave-uniform: EXEC stays all-1s for WMMA

    int half = lane >> 4;   // K-half for A/B fragments, M-half for C/D
    int lr = lane & 15;     // A: row within tile ; B/C/D: column within tile

    // Bias values for all 4 column tiles, hoisted out of the WMMA loops.
    float brv[4], biv[4];
#pragma unroll
    for (int nt = 0; nt < 4; ++nt) {
        brv[nt] = bias_r[nt * 16 + lr];
        biv[nt] = bias_i[nt * 16 + lr];
    }

    // A fragments for all 16 K-chunks, kept in registers (32+32 VGPRs).
    v2f ar[16], ai[16];
    const float* pr = z_real + (size_t)(row0 + lr) * DIM + 2 * half;
    const float* pi = z_imag + (size_t)(row0 + lr) * DIM + 2 * half;
#pragma unroll
    for (int kk = 0; kk < 16; ++kk) {
        ar[kk] = *(const v2f*)(pr + 4 * kk);
        ai[kk] = *(const v2f*)(pi + 4 * kk);
    }

#pragma unroll 1  // keep one cr/ci pair live; full unroll causes scratch spills
    for (int nt = 0; nt < 4; ++nt) {
        int n0 = nt * 16;
        float br = brv[nt];
        float bi = biv[nt];
        v8f cr = {br, br, br, br, br, br, br, br};
        v8f ci = {bi, bi, bi, bi, bi, bi, bi, bi};

#pragma unroll
        for (int kk = 0; kk < 16; ++kk) {
            int boff = (n0 + lr) * DIM + 4 * kk + 2 * half;  // B[k][n] = W[n][k]
            v2f bW1 = *(const v2f*)(&lW[0 * DIM * DIM + boff]);
            v2f bW2 = *(const v2f*)(&lW[1 * DIM * DIM + boff]);
            v2f bP  = *(const v2f*)(&lW[2 * DIM * DIM + boff]);
            v2f bQ  = *(const v2f*)(&lW[3 * DIM * DIM + boff]);
            cr = __builtin_amdgcn_wmma_f32_16x16x4_f32(false, ar[kk], false, bW1,
                                                       (short)0, cr, false, false);
            cr = __builtin_amdgcn_wmma_f32_16x16x4_f32(false, ai[kk], false, bW2,
                                                       (short)0, cr, false, false);
            ci = __builtin_amdgcn_wmma_f32_16x16x4_f32(false, ar[kk], false, bP,
                                                       (short)0, ci, false, false);
            ci = __builtin_amdgcn_wmma_f32_16x16x4_f32(false, ai[kk], false, bQ,
                                                       (short)0, ci, false, false);
        }

#pragma unroll
        for (int v = 0; v < 8; ++v) {
            size_t row = (size_t)(row0 + v + 8 * half);
            out_zr[row * DIM + n0 + lr] = cr[v];
            out_zi[row * DIM + n0 + lr] = ci[v];
        }
    }
}

// ------------------------------------------------------------------- launcher
extern "C" void kernel_launch(void* const* d_in, const int* in_sizes, int n_in,
                              void* d_out, int out_size, void* d_ws, size_t ws_size,
                              hipStream_t stream) {
    const float* h_real = (const float*)d_in[0];
    const float* h_imag = (const float*)d_in[1];
    const float* dnode  = (const float*)d_in[2];
    const float* w_real = (const float*)d_in[3];
    const float* w_imag = (const float*)d_in[4];
    const int*   src    = (const int*)d_in[5];
    const int*   dst    = (const int*)d_in[6];
    const float* W1     = (const float*)d_in[7];
    const float* b1     = (const float*)d_in[8];
    const float* W2     = (const float*)d_in[9];
    const float* b2     = (const float*)d_in[10];

    const int n_nodes = in_sizes[2];
    const int n_edges = in_sizes[3];

    // workspace layout: floats first, then ints
    float* z_real = (float*)d_ws;
    float* z_imag = z_real + (size_t)n_nodes * DIM;
    float* W2m    = z_imag + (size_t)n_nodes * DIM;
    float* P      = W2m + DIM * DIM;
    float* Q      = P + DIM * DIM;
    float* bias_r = Q + DIM * DIM;
    float* bias_i = bias_r + DIM;
    int*   cnt    = (int*)(bias_i + DIM);
    int*   offs   = cnt + n_nodes;          // n_nodes + 1 entries
    int*   cursor = offs + n_nodes + 1;
    int*   eidx   = cursor + n_nodes;       // n_edges entries

    float* out_zr = (float*)d_out;
    float* out_zi = out_zr + (size_t)n_nodes * DIM;

    // 1) zero per-node counts (must be re-zeroed every call)
    sl_zeroi_kernel<<<(n_nodes + 255) / 256, 256, 0, stream>>>(cnt, n_nodes);

    // 2) CSR build: histogram -> scan -> scatter
    sl_count_kernel<<<(n_edges + 255) / 256, 256, 0, stream>>>(dst, cnt, n_edges);
    sl_scan_kernel<<<1, 1024, 0, stream>>>(cnt, offs, cursor, n_nodes);
    sl_scatter_kernel<<<(n_edges + 255) / 256, 256, 0, stream>>>(dst, cursor, eidx, n_edges);

    // 3) derived 64x64 matrices + biases (independent of 1-2; stream-ordered)
    sl_prep_kernel<<<DIM, DIM, 0, stream>>>(W1, W2, b1, b2, W2m, P, Q, bias_r, bias_i);

    // 4) gather-accumulate: one wave per node
    sl_gather_kernel<<<(n_nodes + 7) / 8, 256, 0, stream>>>(h_real, h_imag, dnode,
                                                            w_real, w_imag, src,
                                                            eidx, offs,
                                                            z_real, z_imag, n_nodes);

    // 5) WMMA GEMM: one wave per 16-node tile, 8 tiles per block
    {
        int tiles = (n_nodes + 15) / 16;
        int blocks = (tiles + 7) / 8;
        sl_gemm_kernel<<<blocks, 256, 0, stream>>>(z_real, z_imag, W1, W2m, P, Q,
                                                   bias_r, bias_i, out_zr, out_zi, n_nodes);
    }
}